// RPE_NKA_53678501265930
// MI455X (gfx1250) — compile-verified
//
#include <hip/hip_runtime.h>
#include <hip/hip_bf16.h>
#include <math.h>

// ---------------------------------------------------------------------------
// RPE_NKA faithful-math kernel for MI455X (gfx1250, wave32, WMMA).
//
// out[b',h',f,z] = DFT3(pad(s*v))[b',h',f,z] / DFT3(pad(elu(k/||k||)))[b',h',f,z]
//   k = x@Wk (Frobenius-normalized, elu), v = x@Wv, s = row-sum_d of elu'd k.
//   DFT3 = unnormalized forward DFT over (b:2, h:8, m:1024 zero-padded), f<512.
// q / Wq / Er / exp(skew(QEr)) cancel exactly (see analysis).
// ---------------------------------------------------------------------------

typedef __attribute__((ext_vector_type(2))) float v2f;
typedef __attribute__((ext_vector_type(8))) float v8f;

// workspace layout (float offsets)
#define OFF_WRE   0u            // [512][512] cos
#define OFF_WIM   262144u       // [512][512] -sin
#define OFF_KRAW  524288u       // [1024][512] x@Wk
#define OFF_VMAT  1048576u      // [1024][512] x@Wv
#define OFF_XK    1572864u      // [16][512][64] elu(k/||k||), per (b,h)
#define OFF_XU    2097152u      // [16][512][64] s*v
#define OFF_CKRE  2621440u      // [16][512][64] Re DFT_m(k_)
#define OFF_CKIM  3145728u
#define OFF_CURE  3670016u
#define OFF_CUIM  4194304u
#define OFF_RED   4718592u      // [256] partial sums of squares
#define OFF_SSQ   4718848u      // [1]   total sum of squares

// ---------------------------------------------------------------------------
// f32 WMMA 16x16 tile: C(16x16) += A(16xK) * B(Kx16), row-major, K%4==0.
// A layout (16x4 f32): lanes 0-15 row M=lane, VGPR0/1 = K {0,1}; lanes 16-31 K {2,3}.
// B layout (4x16 f32): lanes 0-15 col N=lane, VGPR0/1 = K {0,1}; lanes 16-31 K {2,3}.
// C layout: VGPR j -> row j (lanes 0-15) / row j+8 (lanes 16-31), col = lane&15.
// ---------------------------------------------------------------------------
__device__ __forceinline__ v8f wmma_f32_tile(const float* __restrict__ A, int lda,
                                             const float* __restrict__ B, int ldb,
                                             int K, int lane)
{
    const int l16 = lane & 15;
    const int kh  = (lane >> 4) << 1;           // 0 or 2
    const float* ar = A + l16 * lda + kh;       // row of A, K-offset kh
    const float* br = B + kh * ldb + l16;       // col of B, K-offset kh
    v8f acc = {};
#pragma unroll 4
    for (int k = 0; k < K; k += 4) {
        v2f a, b;
        a[0] = ar[k];
        a[1] = ar[k + 1];
        b[0] = br[k * ldb];
        b[1] = br[(k + 1) * ldb];
        acc = __builtin_amdgcn_wmma_f32_16x16x4_f32(
            /*neg_a=*/false, a, /*neg_b=*/false, b,
            /*c_mod=*/(short)0, acc, /*reuse_a=*/false, /*reuse_b=*/false);
    }
    return acc;
}

__device__ __forceinline__ void wmma_store_tile(float* __restrict__ C, int ldc,
                                                int lane, v8f acc)
{
    const int ccol = lane & 15;
    const int crow = (lane >> 4) << 3;          // 0 or 8
#pragma unroll
    for (int j = 0; j < 8; ++j)
        C[(crow + j) * ldc + ccol] = acc[j];
}

// ---------------------------------------------------------------------------
// Kernel 0: DFT twiddle matrices W[f][m] = exp(-2*pi*i*f*m/1024), f,m < 512.
// Exact integer phase reduction keeps large f*m accurate.
// ---------------------------------------------------------------------------
__global__ __launch_bounds__(256) void init_w_kernel(float* __restrict__ ws)
{
    const int idx = blockIdx.x * 256 + threadIdx.x;      // 0 .. 262143
    const int f = idx >> 9;
    const int m = idx & 511;
    const int phase = (f * m) & 1023;
    const float ang = (float)phase * 6.1359231515425649e-03f; // 2*pi/1024
    float s, c;
    sincosf(ang, &s, &c);
    ws[OFF_WRE + idx] = c;
    ws[OFF_WIM + idx] = -s;
}

// ---------------------------------------------------------------------------
// Kernel 1: projections kraw = x@Wk (z=0), vmat = x@Wv (z=1).
// M=1024 (b*n), N=512 (h*d), K=512. One wave per 16x16 tile.
// ---------------------------------------------------------------------------
__global__ __launch_bounds__(256) void proj_gemm_kernel(const float* __restrict__ x,
                                                        const float* __restrict__ Wk,
                                                        const float* __restrict__ Wv,
                                                        float* __restrict__ ws)
{
    const float* B = (blockIdx.z == 0) ? Wk : Wv;
    float* C = ws + ((blockIdx.z == 0) ? OFF_KRAW : OFF_VMAT);
    const int wave = (blockIdx.x * 256 + threadIdx.x) >> 5;  // 0 .. 2047
    const int lane = threadIdx.x & 31;
    const int mt = wave >> 5;   // 64 M-tiles
    const int nt = wave & 31;   // 32 N-tiles
    v8f acc = wmma_f32_tile(x + mt * 16 * 512, 512, B + nt * 16, 512, 512, lane);
    wmma_store_tile(C + mt * 16 * 512 + nt * 16, 512, lane, acc);
}

// ---------------------------------------------------------------------------
// Kernel 2a/2b: deterministic Frobenius sum-of-squares of kraw.
// ---------------------------------------------------------------------------
__global__ __launch_bounds__(256) void ssq_partial_kernel(float* __restrict__ ws)
{
    __shared__ float red[256];
    const float* k = ws + OFF_KRAW;
    float acc = 0.f;
    for (int i = blockIdx.x * 256 + threadIdx.x; i < 524288; i += 256 * 256) {
        const float t = k[i];
        acc = fmaf(t, t, acc);
    }
    red[threadIdx.x] = acc;
    __syncthreads();
    for (int s = 128; s > 0; s >>= 1) {
        if ((int)threadIdx.x < s) red[threadIdx.x] += red[threadIdx.x + s];
        __syncthreads();
    }
    if (threadIdx.x == 0) ws[OFF_RED + blockIdx.x] = red[0];
}

__global__ __launch_bounds__(256) void ssq_final_kernel(float* __restrict__ ws)
{
    __shared__ float red[256];
    red[threadIdx.x] = ws[OFF_RED + threadIdx.x];
    __syncthreads();
    for (int s = 128; s > 0; s >>= 1) {
        if ((int)threadIdx.x < s) red[threadIdx.x] += red[threadIdx.x + s];
        __syncthreads();
    }
    if (threadIdx.x == 0) ws[OFF_SSQ] = red[0];
}

// ---------------------------------------------------------------------------
// Kernel 3: per (b,h) repack: Xk = elu(k/||k||), Xu = (sum_d Xk) * v.
// One thread per (b,h,m) row of 64.
// ---------------------------------------------------------------------------
__global__ __launch_bounds__(256) void repack_kernel(float* __restrict__ ws)
{
    const int idx = blockIdx.x * 256 + threadIdx.x;  // 0 .. 8191
    const int bh = idx >> 9;                         // b*8+h
    const int m  = idx & 511;
    const int b = bh >> 3, h = bh & 7;
    const float inv = 1.0f / sqrtf(ws[OFF_SSQ]);
    const float* kr = ws + OFF_KRAW + (unsigned)(b * 512 + m) * 512u + (unsigned)h * 64u;
    const float* vr = ws + OFF_VMAT + (unsigned)(b * 512 + m) * 512u + (unsigned)h * 64u;
    float* xk = ws + OFF_XK + (unsigned)idx * 64u;
    float* xu = ws + OFF_XU + (unsigned)idx * 64u;
    float s = 0.f;
#pragma unroll 8
    for (int d = 0; d < 64; ++d) {
        const float kn = kr[d] * inv;
        const float e = (kn > 0.f) ? kn : expm1f(kn);   // elu, alpha=1
        xk[d] = e;
        s += e;
    }
#pragma unroll 8
    for (int d = 0; d < 64; ++d) xu[d] = s * vr[d];
}

// ---------------------------------------------------------------------------
// Kernel 4: m-axis DFT as WMMA GEMM: C_{re,im}[bh] = W_{re,im} @ X_{k,u}[bh].
// M=512(f) K=512(m) N=64(d). grid.z encodes (tensor, re/im, bh).
// ---------------------------------------------------------------------------
__global__ __launch_bounds__(256) void dft_gemm_kernel(float* __restrict__ ws)
{
    const int z    = blockIdx.z;        // 0..63
    const int bh   = z & 15;
    const int reim = (z >> 4) & 1;
    const int tens = z >> 5;
    const float* A = ws + (reim ? OFF_WIM : OFF_WRE);
    const float* B = ws + (tens ? OFF_XU : OFF_XK) + (unsigned)bh * (512u * 64u);
    float* C = ws + (tens ? (reim ? OFF_CUIM : OFF_CURE)
                          : (reim ? OFF_CKIM : OFF_CKRE)) + (unsigned)bh * (512u * 64u);
    const int wave = (blockIdx.x * 256 + threadIdx.x) >> 5;  // 0 .. 127
    const int lane = threadIdx.x & 31;
    const int mt = wave >> 2;   // 32 f-tiles
    const int nt = wave & 3;    // 4 d-tiles
    v8f acc = wmma_f32_tile(A + mt * 16 * 512, 512, B + nt * 16, 64, 512, lane);
    wmma_store_tile(C + mt * 16 * 64 + nt * 16, 64, lane, acc);
}

// ---------------------------------------------------------------------------
// Kernel 5: 8-point h-DFT, 2-point b-DFT, complex divide, write complex64 out.
// One thread per (f,d); loads all 16 (b,h) values of both tensors.
// ---------------------------------------------------------------------------
__global__ __launch_bounds__(256) void mix_div_kernel(const float* __restrict__ ws,
                                                      float* __restrict__ out)
{
    const int idx = blockIdx.x * 256 + threadIdx.x;     // f*64 + d, 0 .. 32767
    const float* ckre = ws + OFF_CKRE;
    const float* ckim = ws + OFF_CKIM;
    const float* cure = ws + OFF_CURE;
    const float* cuim = ws + OFF_CUIM;

    float kr[16], ki[16], ur[16], ui[16];
#pragma unroll
    for (int bh = 0; bh < 16; ++bh) {
        const unsigned o = (unsigned)bh * (512u * 64u) + (unsigned)idx;
        kr[bh] = ckre[o];
        ki[bh] = ckim[o];
        ur[bh] = cure[o];
        ui[bh] = cuim[o];
    }

    const float R2 = 0.70710678118654752f;   // exp(-2*pi*i*p/8) table
    const float WR[8] = {1.f,  R2, 0.f, -R2, -1.f, -R2, 0.f,  R2};
    const float WI[8] = {0.f, -R2, -1.f, -R2, 0.f,  R2, 1.f,  R2};

#pragma unroll
    for (int hp = 0; hp < 8; ++hp) {
        float Kr[2] = {0.f, 0.f}, Ki[2] = {0.f, 0.f};
        float Ur[2] = {0.f, 0.f}, Ui[2] = {0.f, 0.f};
#pragma unroll
        for (int h = 0; h < 8; ++h) {
            const int p = (hp * h) & 7;
            const float wr = WR[p], wi = WI[p];
#pragma unroll
            for (int b = 0; b < 2; ++b) {
                const int i = b * 8 + h;
                Kr[b] += wr * kr[i] - wi * ki[i];
                Ki[b] += wr * ki[i] + wi * kr[i];
                Ur[b] += wr * ur[i] - wi * ui[i];
                Ui[b] += wr * ui[i] + wi * ur[i];
            }
        }
#pragma unroll
        for (int bp = 0; bp < 2; ++bp) {
            const float sgn = bp ? -1.f : 1.f;
            const float dr = Kr[0] + sgn * Kr[1];
            const float di = Ki[0] + sgn * Ki[1];
            const float nr = Ur[0] + sgn * Ur[1];
            const float ni = Ui[0] + sgn * Ui[1];
            const float den = dr * dr + di * di;
            const float ore = (nr * dr + ni * di) / den;
            const float oim = (ni * dr - nr * di) / den;
            const unsigned o = ((unsigned)(bp * 8 + hp) << 15) + (unsigned)idx;
            out[2u * o]      = ore;
            out[2u * o + 1u] = oim;
        }
    }
}

// ---------------------------------------------------------------------------
extern "C" void kernel_launch(void* const* d_in, const int* in_sizes, int n_in,
                              void* d_out, int out_size, void* d_ws, size_t ws_size,
                              hipStream_t stream) {
    const float* x  = (const float*)d_in[0];   // [2,512,512]
    // d_in[1] = Wq (unused: cancels exactly)
    const float* Wk = (const float*)d_in[2];   // [512,512]
    const float* Wv = (const float*)d_in[3];   // [512,512]
    // d_in[4] = Er (unused: cancels exactly)
    float* ws  = (float*)d_ws;
    float* out = (float*)d_out;                // complex64 interleaved [2,8,512,64]

    init_w_kernel<<<1024, 256, 0, stream>>>(ws);
    proj_gemm_kernel<<<dim3(256, 1, 2), 256, 0, stream>>>(x, Wk, Wv, ws);
    ssq_partial_kernel<<<256, 256, 0, stream>>>(ws);
    ssq_final_kernel<<<1, 256, 0, stream>>>(ws);
    repack_kernel<<<32, 256, 0, stream>>>(ws);
    dft_gemm_kernel<<<dim3(16, 1, 64), 256, 0, stream>>>(ws);
    mix_div_kernel<<<128, 256, 0, stream>>>(ws, out);
}